// NewtinImplictNet_34840774705291
// MI455X (gfx1250) — compile-verified
//
#include <hip/hip_runtime.h>

// ---------------------------------------------------------------------------
// NewtinImplictNet on MI455X (gfx1250), fp32 WMMA path.
//
// z = x@w_in + b_in
// 4x: c = z + tanh(z) + b ; z = c@W^-1 (Newton converges in 1 step for a
//     linear system) ; refinement z = z + c - z@W  (reference's final step)
// out = softmax(z@w_out + b_out)
//
// W^-1 computed on-device with Newton-Schulz (X <- X(2I - W X), X0 = 2I - W),
// quadratic convergence to fp32 exactness in 4 iterations, batched over the
// 4 implicit-layer matrices, using the same WMMA GEMM kernel.
// ---------------------------------------------------------------------------

typedef float v2f __attribute__((ext_vector_type(2)));
typedef float v8f __attribute__((ext_vector_type(8)));

#define NROWS   65536
#define DIM     256
#define ODIM    10

#define BM 128
#define BN 128
#define BK 32
#define PADA 4   // A row pad: (BK+4)*4 = 144B, multiple of 16 -> aligned b128 stores

// ---------------------------------------------------------------------------
// Tiled fp32 GEMM: D = epi(A @ Beff), A: M x K row-major (lda=K),
// B: K x N row-major (ldb=N). 256 threads = 8 waves; waves in 4(M) x 2(N);
// per-wave 32x64 = 2x4 grid of 16x16 WMMA accumulators.
// A staged m-major (k contiguous) -> A-frag is one b64 load.
// B staged k-pair interleaved Bs[kp][n][2] -> B-frag is one b64 load into an
// even-aligned VGPR pair (no repack movs before v_wmma).
// BMODE: 0 -> Beff = B ; 1 -> Beff = 2I - B   (Newton-Schulz fuse)
// EPI:   0 -> D = acc ; 1 -> D = acc + bias[n] ; 2 -> D = aux1 + aux2 - acc
// Batched via blockIdx.z with element strides sA/sB/sD.
// ---------------------------------------------------------------------------
template <int BMODE, int EPI>
__global__ __launch_bounds__(256) void gemm_wmma_f32(
    const float* __restrict__ A, long sA,
    const float* __restrict__ B, long sB,
    float* __restrict__ D, long sD,
    const float* __restrict__ bias,
    const float* __restrict__ aux1,
    const float* __restrict__ aux2,
    int M, int N, int K) {
  __shared__ float As[BM][BK + PADA];        // m-major, k contiguous
  __shared__ float Bs[BK / 2][BN + 1][2];    // {k even, k odd} interleaved

  const int tid = threadIdx.x;
  const long bz = blockIdx.z;
  A += bz * sA;
  B += bz * sB;
  D += bz * sD;

  const int nBase = blockIdx.x * BN;
  const int mBase = blockIdx.y * BM;

  const int wave = tid >> 5;
  const int lane = tid & 31;
  const int half = lane >> 4;   // K-half select (ISA 16x4 f32 A layout)
  const int lr   = lane & 15;   // M (A) / N (B,C) index within frag
  const int waveM = (wave & 3) * 32;
  const int waveN = (wave >> 2) * 64;

  v8f acc[2][4];
#pragma unroll
  for (int mt = 0; mt < 2; ++mt)
#pragma unroll
    for (int nt = 0; nt < 4; ++nt) acc[mt][nt] = (v8f)0.0f;

  for (int k0 = 0; k0 < K; k0 += BK) {
    // --- stage A tile [BM x BK] ---
#pragma unroll
    for (int it = 0; it < 4; ++it) {
      int idx = it * 256 + tid;          // over 128*8 float4 slots
      int m = idx >> 3;
      int k4 = (idx & 7) << 2;
      float4 v = *(const float4*)(A + (long)(mBase + m) * K + (k0 + k4));
      *(float4*)(&As[m][k4]) = v;
    }
    // --- stage B tile [BK x BN], k-pair interleaved ---
#pragma unroll
    for (int it = 0; it < 4; ++it) {
      int idx = it * 256 + tid;          // over 32*32 float4 slots
      int k = idx >> 5;
      int n4 = (idx & 31) << 2;
      int kp = k >> 1, ko = k & 1;
      if (BMODE == 0) {
        float4 v = *(const float4*)(B + (long)(k0 + k) * N + (nBase + n4));
        Bs[kp][n4 + 0][ko] = v.x;
        Bs[kp][n4 + 1][ko] = v.y;
        Bs[kp][n4 + 2][ko] = v.z;
        Bs[kp][n4 + 3][ko] = v.w;
      } else {  // Beff = 2I - B
        int gk = k0 + k;
#pragma unroll
        for (int u = 0; u < 4; ++u) {
          int gn = nBase + n4 + u;
          float w = B[(long)gk * N + gn];
          Bs[kp][n4 + u][ko] = (gk == gn ? 2.0f : 0.0f) - w;
        }
      }
    }
    __syncthreads();

#pragma unroll
    for (int kc = 0; kc < BK; kc += 4) {
      const int ks = kc + half * 2;       // this lane-half's K pair (A side)
      const int kp = (kc >> 1) + half;    // interleaved pair index (B side)
      v2f af[2], bf[4];
#pragma unroll
      for (int mt = 0; mt < 2; ++mt) {
        int m = waveM + mt * 16 + lr;
        af[mt] = *(const v2f*)(&As[m][ks]);        // one b64
      }
#pragma unroll
      for (int nt = 0; nt < 4; ++nt) {
        int n = waveN + nt * 16 + lr;
        bf[nt] = *(const v2f*)(&Bs[kp][n][0]);     // one b64, even-pair ready
      }
#pragma unroll
      for (int mt = 0; mt < 2; ++mt)
#pragma unroll
        for (int nt = 0; nt < 4; ++nt)
          acc[mt][nt] = __builtin_amdgcn_wmma_f32_16x16x4_f32(
              false, af[mt], false, bf[nt], (short)0, acc[mt][nt], false, false);
    }
    __syncthreads();
  }

  // --- epilogue: C/D layout: VGPR r -> M = r + 8*half, lanes -> N = lr ---
#pragma unroll
  for (int mt = 0; mt < 2; ++mt) {
#pragma unroll
    for (int nt = 0; nt < 4; ++nt) {
      int col = nBase + waveN + nt * 16 + lr;
#pragma unroll
      for (int r = 0; r < 8; ++r) {
        int row = mBase + waveM + mt * 16 + half * 8 + r;
        long off = (long)row * N + col;
        float v = acc[mt][nt][r];
        if (EPI == 0)
          D[off] = v;
        else if (EPI == 1)
          D[off] = v + bias[col];
        else
          D[off] = aux1[off] + aux2[off] - v;  // z + c - z@W
      }
    }
  }
}

// Pack the 4 implicit-layer weights contiguously and seed X0 = 2I - W.
__global__ __launch_bounds__(256) void pack_init(
    const float* __restrict__ w0, const float* __restrict__ w1,
    const float* __restrict__ w2, const float* __restrict__ w3,
    float* __restrict__ Wc, float* __restrict__ Xa) {
  int z = blockIdx.z;
  const float* W = (z == 0) ? w0 : (z == 1) ? w1 : (z == 2) ? w2 : w3;
  int idx = blockIdx.x * 256 + threadIdx.x;  // grid.x = 256 -> 65536 elements
  float w = W[idx];
  int i = idx >> 8, j = idx & 255;
  Wc[(long)z * (DIM * DIM) + idx] = w;
  Xa[(long)z * (DIM * DIM) + idx] = (i == j ? 2.0f : 0.0f) - w;
}

// c = z + tanh(z) + b  (float4 vectorized; DIM divides 4)
__global__ __launch_bounds__(256) void make_c(const float* __restrict__ Z,
                                              const float* __restrict__ b,
                                              float* __restrict__ C) {
  long idx4 = (long)blockIdx.x * 256 + threadIdx.x;
  long base = idx4 * 4;
  int n = (int)(base & (DIM - 1));
  float4 z = *(const float4*)(Z + base);
  float4 c;
  c.x = z.x + tanhf(z.x) + b[n + 0];
  c.y = z.y + tanhf(z.y) + b[n + 1];
  c.z = z.z + tanhf(z.z) + b[n + 2];
  c.w = z.w + tanhf(z.w) + b[n + 3];
  *(float4*)(C + base) = c;
}

// logits = z @ w_out + b_out ; softmax rows. One wave (32 lanes) per row.
__global__ __launch_bounds__(256) void out_softmax(
    const float* __restrict__ Z, const float* __restrict__ wout,
    const float* __restrict__ bout, float* __restrict__ out) {
  __shared__ float sw[DIM * ODIM];
  __shared__ float sb[ODIM];
  int tid = threadIdx.x;
  for (int i = tid; i < DIM * ODIM; i += 256) sw[i] = wout[i];
  if (tid < ODIM) sb[tid] = bout[tid];
  __syncthreads();

  int wave = tid >> 5, lane = tid & 31;
  long row = (long)blockIdx.x * 8 + wave;
  const float* zr = Z + row * DIM + lane * 8;
  float4 z0 = *(const float4*)zr;
  float4 z1 = *(const float4*)(zr + 4);
  float zz[8] = {z0.x, z0.y, z0.z, z0.w, z1.x, z1.y, z1.z, z1.w};

  float p[ODIM];
#pragma unroll
  for (int j = 0; j < ODIM; ++j) p[j] = 0.0f;
#pragma unroll
  for (int q = 0; q < 8; ++q) {
    int k = lane * 8 + q;
    const float* wk = &sw[k * ODIM];
#pragma unroll
    for (int j = 0; j < ODIM; ++j) p[j] = fmaf(zz[q], wk[j], p[j]);
  }
#pragma unroll
  for (int off = 16; off >= 1; off >>= 1)
#pragma unroll
    for (int j = 0; j < ODIM; ++j) p[j] += __shfl_xor(p[j], off, 32);

  if (lane == 0) {
    float mx = -3.4e38f;
#pragma unroll
    for (int j = 0; j < ODIM; ++j) {
      p[j] += sb[j];
      mx = fmaxf(mx, p[j]);
    }
    float s = 0.0f;
#pragma unroll
    for (int j = 0; j < ODIM; ++j) {
      p[j] = __expf(p[j] - mx);
      s += p[j];
    }
    float inv = 1.0f / s;
#pragma unroll
    for (int j = 0; j < ODIM; ++j) out[row * ODIM + j] = p[j] * inv;
  }
}

// ---------------------------------------------------------------------------
extern "C" void kernel_launch(void* const* d_in, const int* in_sizes, int n_in,
                              void* d_out, int out_size, void* d_ws, size_t ws_size,
                              hipStream_t stream) {
  (void)in_sizes; (void)n_in; (void)out_size; (void)ws_size;
  const float* x     = (const float*)d_in[0];
  const float* w_in  = (const float*)d_in[1];
  const float* b_in  = (const float*)d_in[2];
  const float* w_out = (const float*)d_in[3];
  const float* b_out = (const float*)d_in[4];
  const float* W[4]  = {(const float*)d_in[5], (const float*)d_in[7],
                        (const float*)d_in[9], (const float*)d_in[11]};
  const float* bb[4] = {(const float*)d_in[6], (const float*)d_in[8],
                        (const float*)d_in[10], (const float*)d_in[12]};

  const long NZ = (long)NROWS * DIM;     // 16.7M floats
  const long WSZ = (long)DIM * DIM;      // 65536
  float* ws = (float*)d_ws;
  float* Z  = ws;            // 64 MB
  float* C  = Z + NZ;        // 64 MB
  float* T  = C + NZ;        // 64 MB
  float* Xa = T + NZ;        // 1 MB  (inverse iterate / result)
  float* Xb = Xa + 4 * WSZ;  // 1 MB
  float* Tm = Xb + 4 * WSZ;  // 1 MB  (Newton-Schulz temp)
  float* Wc = Tm + 4 * WSZ;  // 1 MB  (packed W1..W4)

  dim3 blk(256);

  // --- pack weights, seed X0 = 2I - W ---
  pack_init<<<dim3(256, 1, 4), blk, 0, stream>>>(W[0], W[1], W[2], W[3], Wc, Xa);

  // --- Newton-Schulz: X <- X (2I - W X), 4 iters, batched over 4 matrices ---
  float* Xc = Xa; float* Xn = Xb;
  for (int it = 0; it < 4; ++it) {
    gemm_wmma_f32<0, 0><<<dim3(2, 2, 4), blk, 0, stream>>>(
        Wc, WSZ, Xc, WSZ, Tm, WSZ, nullptr, nullptr, nullptr, DIM, DIM, DIM);
    gemm_wmma_f32<1, 0><<<dim3(2, 2, 4), blk, 0, stream>>>(
        Xc, WSZ, Tm, WSZ, Xn, WSZ, nullptr, nullptr, nullptr, DIM, DIM, DIM);
    float* t = Xc; Xc = Xn; Xn = t;
  }
  // even number of swaps: Xc == Xa holds W^-1 for the 4 layers

  // --- input layer: Z = x @ w_in + b_in ---
  gemm_wmma_f32<0, 1><<<dim3(DIM / BN, NROWS / BM, 1), blk, 0, stream>>>(
      x, 0, w_in, 0, Z, 0, b_in, nullptr, nullptr, NROWS, DIM, DIM);

  // --- 4 implicit blocks ---
  for (int i = 0; i < 4; ++i) {
    make_c<<<dim3((unsigned)(NZ / 4 / 256)), blk, 0, stream>>>(Z, bb[i], C);
    // T = C @ W^-1   (Newton converged solve)
    gemm_wmma_f32<0, 0><<<dim3(DIM / BN, NROWS / BM, 1), blk, 0, stream>>>(
        C, 0, Xc + (long)i * WSZ, 0, T, 0, nullptr, nullptr, nullptr, NROWS, DIM, DIM);
    // Z = T + C - T@W   (reference's final differentiable refinement)
    gemm_wmma_f32<0, 2><<<dim3(DIM / BN, NROWS / BM, 1), blk, 0, stream>>>(
        T, 0, Wc + (long)i * WSZ, 0, Z, 0, nullptr, T, C, NROWS, DIM, DIM);
  }

  // --- output layer + softmax ---
  out_softmax<<<dim3(NROWS / 8), blk, 0, stream>>>(Z, w_out, b_out, (float*)d_out);
}